// CombinedBoundaryDiceLoss_44762149159188
// MI455X (gfx1250) — compile-verified
//
#include <hip/hip_runtime.h>
#include <hip/hip_bf16.h>
#include <math.h>

#define B_   16
#define H_   512
#define W_   512
#define N_   (B_*H_*W_)      /* 4194304 */
#define BIGF 1000000.0f
#define ALPHA_  0.5f
#define BETA_   0.5f
#define SMOOTH_ 1e-6f

typedef __attribute__((ext_vector_type(2))) float v2f;
typedef __attribute__((ext_vector_type(8))) float v8f;

// ---------------------------------------------------------------------------
// Kernel 1: per-row 1D distance (forward + backward scan), write g^2 for both
// masks (fg: t>0.5, bg: t<=0.5). One thread per (b,h) row.
// ---------------------------------------------------------------------------
__global__ void rowdist_kernel(const int* __restrict__ tgt,
                               float* __restrict__ g2fg,
                               float* __restrict__ g2bg) {
  int r = blockIdx.x * blockDim.x + threadIdx.x;   // row id in [0, B*H)
  if (r >= B_ * H_) return;
  const long base = (long)r * W_;

  // forward scan: c = mask ? 0 : c+1  (matches lax.scan with init=BIG)
  float cf = BIGF, cb = BIGF;
  for (int w = 0; w < W_; ++w) {
    int t = tgt[base + w];
    bool mf = (t != 0);
    cf = mf ? 0.0f : cf + 1.0f;
    cb = mf ? cb + 1.0f : 0.0f;
    g2fg[base + w] = cf;          // temporarily store fwd distance
    g2bg[base + w] = cb;
  }
  // backward scan, combine, square
  cf = BIGF; cb = BIGF;
  for (int w = W_ - 1; w >= 0; --w) {
    int t = tgt[base + w];
    bool mf = (t != 0);
    cf = mf ? 0.0f : cf + 1.0f;
    cb = mf ? cb + 1.0f : 0.0f;
    float gf = fminf(g2fg[base + w], cf);
    float gb = fminf(g2bg[base + w], cb);
    g2fg[base + w] = gf * gf;
    g2bg[base + w] = gb * gb;
  }
}

// ---------------------------------------------------------------------------
// Kernel 2 (hot): column min-plus pass, fused with sigmoid + loss reductions.
// Block = (b, 32-column tile). LDS holds the full 512-row column tile of both
// g2 arrays, staged with CDNA5 async-to-LDS loads (ASYNCcnt path).
// Each thread owns 8 'i' accumulators per mask so one LDS load feeds 16
// fma+min ops. (i-i')^2 is exact in f32 => fma(t,t,g2) matches reference.
// ---------------------------------------------------------------------------
__global__ __launch_bounds__(256)
void coldist_fused_kernel(const float* __restrict__ inputs,
                          const int*   __restrict__ tgt,
                          const float* __restrict__ g2fg,
                          const float* __restrict__ g2bg,
                          float*       __restrict__ partials) {
  __shared__ __align__(16) float tileFg[H_ * 32];   // 64 KB
  __shared__ __align__(16) float tileBg[H_ * 32];   // 64 KB

  const int tid = threadIdx.x;
  const int b   = blockIdx.x >> 4;     // /16
  const int j0  = (blockIdx.x & 15) * 32;

  // ---- async staging: 512x32 floats per mask, 16B per lane per instruction
  {
    unsigned ldsFg = (unsigned)(size_t)(&tileFg[0]);
    unsigned ldsBg = (unsigned)(size_t)(&tileBg[0]);
    for (int e = tid; e < (H_ * 32) / 4; e += 256) {   // 16 iterations
      int i  = e >> 3;                 // row (8 x 16B chunks per 128B row)
      int c4 = e & 7;
      long goff = (((long)(b * H_ + i) * W_) + j0 + c4 * 4) * 4;  // bytes
      unsigned loff = (unsigned)(e * 16);
      const char* gf = (const char*)g2fg + goff;
      const char* gb = (const char*)g2bg + goff;
      unsigned lf = ldsFg + loff;
      unsigned lb = ldsBg + loff;
      asm volatile("global_load_async_to_lds_b128 %0, %1, off"
                   :: "v"(lf), "v"(gf) : "memory");
      asm volatile("global_load_async_to_lds_b128 %0, %1, off"
                   :: "v"(lb), "v"(gb) : "memory");
    }
#if __has_builtin(__builtin_amdgcn_s_wait_asynccnt)
    __builtin_amdgcn_s_wait_asynccnt(0);
#else
    asm volatile("s_wait_asynccnt 0" ::: "memory");
#endif
    __syncthreads();
  }

  const int jj = tid & 31;     // column within tile (lane)
  const int w8 = tid >> 5;     // wave id 0..7

  float S1 = 0.0f, S2 = 0.0f, S3 = 0.0f, S4 = 0.0f;

  for (int c = 0; c < 8; ++c) {
    const int ibase = c * 64 + w8 * 8;       // 8 consecutive output rows
    float mfg[8], mbg[8];
#pragma unroll
    for (int k = 0; k < 8; ++k) { mfg[k] = 3.4e38f; mbg[k] = 3.4e38f; }

    float d0 = (float)ibase;                 // ibase - ip, updated per iter
    for (int ip = 0; ip < H_; ++ip) {
      float gf = tileFg[ip * 32 + jj];
      float gb = tileBg[ip * 32 + jj];
#pragma unroll
      for (int k = 0; k < 8; ++k) {
        float t = d0 + (float)k;             // integer-valued, t*t exact
        mfg[k] = fminf(mfg[k], __builtin_fmaf(t, t, gf));
        mbg[k] = fminf(mbg[k], __builtin_fmaf(t, t, gb));
      }
      d0 -= 1.0f;
    }

#pragma unroll
    for (int k = 0; k < 8; ++k) {
      int  i    = ibase + k;
      float dist = sqrtf(mfg[k]) + sqrtf(mbg[k]);
      long idx  = ((long)(b * H_ + i) * W_) + j0 + jj;
      float x   = inputs[idx];
      float p   = 1.0f / (1.0f + __expf(-x));
      float tf  = (tgt[idx] != 0) ? 1.0f : 0.0f;
      S1 += p * dist;
      S2 += p;
      S3 += p * tf;
      S4 += tf;
    }
  }

  // ---- deterministic block tree reduction (reuse tile LDS) ----
  __syncthreads();
  float4* red = (float4*)tileFg;
  red[tid] = make_float4(S1, S2, S3, S4);
  __syncthreads();
  for (int off = 128; off > 0; off >>= 1) {
    if (tid < off) {
      float4 a = red[tid], c4 = red[tid + off];
      red[tid] = make_float4(a.x + c4.x, a.y + c4.y, a.z + c4.z, a.w + c4.w);
    }
    __syncthreads();
  }
  if (tid == 0) {
    float4 r = red[0];
    partials[0 * 256 + blockIdx.x] = r.x;
    partials[1 * 256 + blockIdx.x] = r.y;
    partials[2 * 256 + blockIdx.x] = r.z;
    partials[3 * 256 + blockIdx.x] = r.w;
  }
}

// ---------------------------------------------------------------------------
// Kernel 3: 256 -> 1 reduction per sum using V_WMMA_F32_16X16X4_F32 with
// A = ones (tensor-core column-sum; grand total is invariant to the exact
// B-matrix VGPR mapping), then a 16-lane shuffle finish. Single wave.
// ---------------------------------------------------------------------------
__global__ void finalize_kernel(const float* __restrict__ partials,
                                float* __restrict__ out) {
  int l = threadIdx.x;   // 0..31, one wave, EXEC all ones
  float tot[4];

#if __has_builtin(__builtin_amdgcn_wmma_f32_16x16x4_f32)
  v2f a; a[0] = 1.0f; a[1] = 1.0f;       // A(16x4) = ones
  int i0 = (l < 16) ? l      : l + 16;   // B[k,n] <- partials[16k+n]
  int i1 = (l < 16) ? l + 16 : l + 32;
  for (int s = 0; s < 4; ++s) {
    const float* base = partials + s * 256;
    v8f cacc = {0.0f, 0.0f, 0.0f, 0.0f, 0.0f, 0.0f, 0.0f, 0.0f};
#pragma unroll
    for (int ch = 0; ch < 4; ++ch) {     // 4 chunks x 64 values
      v2f bvec;
      bvec[0] = base[ch * 64 + i0];
      bvec[1] = base[ch * 64 + i1];
      cacc = __builtin_amdgcn_wmma_f32_16x16x4_f32(
          false, a, false, bvec, (short)0, cacc, false, false);
    }
    float v = cacc[0];                   // lanes 0..15 hold the 16 colsums
    v += __shfl_xor(v, 8, 32);
    v += __shfl_xor(v, 4, 32);
    v += __shfl_xor(v, 2, 32);
    v += __shfl_xor(v, 1, 32);
    tot[s] = v;
  }
#else
  for (int s = 0; s < 4; ++s) {
    const float* base = partials + s * 256;
    float v = 0.0f;
    for (int k = 0; k < 8; ++k) v += base[k * 32 + l];
    v += __shfl_xor(v, 16, 32);
    v += __shfl_xor(v, 8, 32);
    v += __shfl_xor(v, 4, 32);
    v += __shfl_xor(v, 2, 32);
    v += __shfl_xor(v, 1, 32);
    tot[s] = v;
  }
#endif

  if (l == 0) {
    float S1 = tot[0], S2 = tot[1], S3 = tot[2], S4 = tot[3];
    float loss_b = BETA_ * (S1 / (float)N_);
    float dice   = (2.0f * S3 + SMOOTH_) / (S2 + S4 + SMOOTH_);
    float loss_d = 1.0f - dice;
    out[0] = ALPHA_ * loss_d + BETA_ * loss_b;   // note: BETA applied twice,
  }                                              // exactly as in reference
}

// ---------------------------------------------------------------------------
extern "C" void kernel_launch(void* const* d_in, const int* in_sizes, int n_in,
                              void* d_out, int out_size, void* d_ws, size_t ws_size,
                              hipStream_t stream) {
  (void)in_sizes; (void)n_in; (void)out_size; (void)ws_size;
  const float* inputs  = (const float*)d_in[0];
  const int*   targets = (const int*)d_in[1];

  // workspace: g2fg[N], g2bg[N], partials[4*256]  (~33.6 MB)
  float* g2fg     = (float*)d_ws;
  float* g2bg     = g2fg + N_;
  float* partials = g2bg + N_;

  rowdist_kernel<<<(B_ * H_ + 255) / 256, 256, 0, stream>>>(targets, g2fg, g2bg);
  coldist_fused_kernel<<<B_ * (W_ / 32), 256, 0, stream>>>(inputs, targets,
                                                           g2fg, g2bg, partials);
  finalize_kernel<<<1, 32, 0, stream>>>(partials, (float*)d_out);
}